// GNN_1185410974040
// MI455X (gfx1250) — compile-verified
//
#include <hip/hip_runtime.h>
#include <hip/hip_bf16.h>

typedef __attribute__((ext_vector_type(2))) float v2f;
typedef __attribute__((ext_vector_type(8))) float v8f;

#define HSTRIDE 64   // padded feature stride (50 used, rest zero)
#define GSTRIDE 52   // pooled-sum stride
#define NGRAPH  128

// ---------------- utility ----------------
__global__ void zero_f32(float* __restrict__ p, int n) {
  int i = blockIdx.x * blockDim.x + threadIdx.x;
  if (i < n) p[i] = 0.0f;
}

// ---------------- degree (in-degree, self-loop added later) ----------------
__global__ void deg_kernel(const int* __restrict__ dst, float* __restrict__ deg, int ne) {
  int e = blockIdx.x * blockDim.x + threadIdx.x;
  if (e < ne) atomicAdd(&deg[dst[e]], 1.0f);
}

// dinv = rsqrt(deg+1);  s = dinv^2 * x   (self-loop term of layer-1 scalar agg)
__global__ void dinv_s_init(const float* __restrict__ deg, const float* __restrict__ x,
                            float* __restrict__ dinv, float* __restrict__ s, int n) {
  int i = blockIdx.x * blockDim.x + threadIdx.x;
  if (i >= n) return;
  float dv = rsqrtf(deg[i] + 1.0f);
  dinv[i] = dv;
  s[i] = dv * dv * x[i];
}

// layer-1 scalar aggregation: s[dst] += dinv[src]*dinv[dst]*x[src]
__global__ void edge_s(const int* __restrict__ src, const int* __restrict__ dst,
                       const float* __restrict__ dinv, const float* __restrict__ x,
                       float* __restrict__ s, int ne) {
  int e = blockIdx.x * blockDim.x + threadIdx.x;
  if (e >= ne) return;
  int sc = src[e], d = dst[e];
  atomicAdd(&s[d], dinv[sc] * dinv[d] * x[sc]);
}

// h1 = relu(s*W1 + b1) (padded to 64 with zeros);
// a  = dinv^2 * h1  (self-loop term of layer-2 aggregation, also zero-pads a);
// also count nodes per graph.
__global__ void h1_a_init(const float* __restrict__ s, const float* __restrict__ dinv,
                          const float* __restrict__ W1, const float* __restrict__ b1,
                          const int* __restrict__ batch,
                          float* __restrict__ h1, float* __restrict__ a,
                          float* __restrict__ gcnt, int n) {
  int gid = blockIdx.x * blockDim.x + threadIdx.x;
  int node = gid >> 6;
  int j = gid & 63;
  if (node >= n) return;
  float h = 0.0f;
  if (j < 50) h = fmaxf(fmaf(s[node], W1[j], b1[j]), 0.0f);
  size_t off = (size_t)node * HSTRIDE + j;
  h1[off] = h;
  float dv = dinv[node];
  a[off] = dv * dv * h;
  if (j == 0) atomicAdd(&gcnt[batch[node]], 1.0f);
}

// layer-2 edge aggregation over 50-wide features: 16 lanes per edge
__global__ void edge_agg(const int* __restrict__ src, const int* __restrict__ dst,
                         const float* __restrict__ dinv, const float* __restrict__ h1,
                         float* __restrict__ a, int ne) {
  int gid = blockIdx.x * blockDim.x + threadIdx.x;
  int e = gid >> 4;
  int j = gid & 15;
  if (e >= ne) return;
  int sidx = src[e], didx = dst[e];
  float w = dinv[sidx] * dinv[didx];
  const float* __restrict__ hs = h1 + (size_t)sidx * HSTRIDE;
  float* __restrict__ ad = a + (size_t)didx * HSTRIDE;
#pragma unroll
  for (int c = j; c < 50; c += 16) atomicAdd(&ad[c], w * hs[c]);
}

// WMMA GEMM: [N,50(pad 52)] x [50,50] with fused bias+ReLU+mean-pool atomics.
// One wave per 16-row M-tile; 4 column tiles (N padded to 64).
// f32 WMMA 16x16x4 per ISA 7.12.2 layouts.
__global__ __launch_bounds__(256) void gemm_relu_pool(
    const float* __restrict__ a, const float* __restrict__ W2,
    const float* __restrict__ b2, const int* __restrict__ batch,
    float* __restrict__ gsum, int n) {
  // B in LDS, paired-k layout: BsP[k>>1][col 0..63][k&1] so (k,k+1) are a b64 load
  __shared__ float BsP[52 * 64];
  for (int i = threadIdx.x; i < 52 * 64; i += 256) {
    int k = i >> 6, c = i & 63;
    float v = (k < 50 && c < 50) ? W2[k * 50 + c] : 0.0f;
    BsP[((k >> 1) * 64 + c) * 2 + (k & 1)] = v;
  }
  __syncthreads();

  const int wave = threadIdx.x >> 5;
  const int lane = threadIdx.x & 31;
  const int col  = lane & 15;
  const int kk   = (lane >> 4) << 1;  // 0 or 2: K offset for A/B fragments
  const int hi   = (lane >> 4) << 3;  // 0 or 8: row offset for C/D epilogue

  int m0 = (blockIdx.x * 8 + wave) * 16;
  if (m0 >= n) return;

  int rowA = m0 + col;
  // clamp OOB rows to row 0: their products are never stored, values irrelevant
  const float* __restrict__ arow = a + (size_t)(rowA < n ? rowA : 0) * HSTRIDE;

  v8f acc0 = {}, acc1 = {}, acc2 = {}, acc3 = {};
#pragma unroll
  for (int k0 = 0; k0 < 52; k0 += 4) {
    int k = k0 + kk;                       // even
    v2f af = *(const v2f*)(arow + k);      // A[m, k], A[m, k+1]  (b64 load)
    const float* bp = &BsP[((k >> 1) * 64) * 2];
    v2f bf;
    bf = *(const v2f*)(bp + 2 * (col));
    acc0 = __builtin_amdgcn_wmma_f32_16x16x4_f32(false, af, false, bf, (short)0, acc0, false, false);
    bf = *(const v2f*)(bp + 2 * (16 + col));
    acc1 = __builtin_amdgcn_wmma_f32_16x16x4_f32(false, af, false, bf, (short)0, acc1, false, false);
    bf = *(const v2f*)(bp + 2 * (32 + col));
    acc2 = __builtin_amdgcn_wmma_f32_16x16x4_f32(false, af, false, bf, (short)0, acc2, false, false);
    bf = *(const v2f*)(bp + 2 * (48 + col));
    acc3 = __builtin_amdgcn_wmma_f32_16x16x4_f32(false, af, false, bf, (short)0, acc3, false, false);
  }

  // epilogue: bias + relu + pooled atomic add
  float bias0 = b2[col];
  float bias1 = b2[16 + col];
  float bias2 = b2[32 + col];
  float bias3 = (48 + col < 50) ? b2[48 + col] : 0.0f;
#pragma unroll
  for (int g = 0; g < 8; ++g) {
    int row = m0 + g + hi;
    if (row >= n) continue;
    float* __restrict__ gs = gsum + (size_t)batch[row] * GSTRIDE;
    float v;
    v = fmaxf(acc0[g] + bias0, 0.0f); atomicAdd(&gs[col], v);
    v = fmaxf(acc1[g] + bias1, 0.0f); atomicAdd(&gs[16 + col], v);
    v = fmaxf(acc2[g] + bias2, 0.0f); atomicAdd(&gs[32 + col], v);
    if (48 + col < 50) { v = fmaxf(acc3[g] + bias3, 0.0f); atomicAdd(&gs[48 + col], v); }
  }
}

// out[g,c] = (gsum[g]/max(cnt,1)) . Wc[:,c] + bc[c] ; 128 graphs x 2 classes
__global__ void classifier(const float* __restrict__ gsum, const float* __restrict__ gcnt,
                           const float* __restrict__ Wc, const float* __restrict__ bc,
                           float* __restrict__ out) {
  int t = threadIdx.x;
  if (t >= NGRAPH * 2) return;
  int g = t >> 1, c = t & 1;
  float inv = 1.0f / fmaxf(gcnt[g], 1.0f);
  float acc = bc[c];
  for (int j = 0; j < 50; ++j)
    acc = fmaf(gsum[g * GSTRIDE + j] * inv, Wc[j * 2 + c], acc);
  out[g * 2 + c] = acc;
}

extern "C" void kernel_launch(void* const* d_in, const int* in_sizes, int n_in,
                              void* d_out, int out_size, void* d_ws, size_t ws_size,
                              hipStream_t stream) {
  const float* x     = (const float*)d_in[0];
  const int*   ei    = (const int*)d_in[1];
  const int*   batch = (const int*)d_in[2];
  const float* W1    = (const float*)d_in[3];
  const float* b1    = (const float*)d_in[4];
  const float* W2    = (const float*)d_in[5];
  const float* b2    = (const float*)d_in[6];
  const float* Wc    = (const float*)d_in[7];
  const float* bc    = (const float*)d_in[8];
  float* out = (float*)d_out;

  int n  = in_sizes[0];       // x is [N,1]
  int ne = in_sizes[1] / 2;   // edge_index is [2,E]
  const int* src = ei;
  const int* dst = ei + ne;

  float* ws   = (float*)d_ws;
  float* deg  = ws;  ws += n;
  float* dinv = ws;  ws += n;
  float* s    = ws;  ws += n;
  float* h1   = ws;  ws += (size_t)n * HSTRIDE;
  float* a    = ws;  ws += (size_t)n * HSTRIDE;
  float* gsum = ws;  ws += NGRAPH * GSTRIDE;
  float* gcnt = ws;  ws += NGRAPH;

  const int B = 256;
  zero_f32<<<(n + B - 1) / B, B, 0, stream>>>(deg, n);
  zero_f32<<<(NGRAPH * GSTRIDE + NGRAPH + B - 1) / B, B, 0, stream>>>(gsum, NGRAPH * GSTRIDE + NGRAPH);
  deg_kernel<<<(ne + B - 1) / B, B, 0, stream>>>(dst, deg, ne);
  dinv_s_init<<<(n + B - 1) / B, B, 0, stream>>>(deg, x, dinv, s, n);
  edge_s<<<(ne + B - 1) / B, B, 0, stream>>>(src, dst, dinv, x, s, ne);
  h1_a_init<<<(int)(((size_t)n * 64 + B - 1) / B), B, 0, stream>>>(s, dinv, W1, b1, batch, h1, a, gcnt, n);
  long long agg_threads = (long long)ne * 16;
  edge_agg<<<(int)((agg_threads + B - 1) / B), B, 0, stream>>>(src, dst, dinv, h1, a, ne);
  int mtiles = (n + 15) / 16;
  gemm_relu_pool<<<(mtiles + 7) / 8, B, 0, stream>>>(a, W2, b2, batch, gsum, n);
  classifier<<<1, 256, 0, stream>>>(gsum, gcnt, Wc, bc, out);
}